// NgramGPUKernel_13709535609523
// MI455X (gfx1250) — compile-verified
//
#include <hip/hip_runtime.h>
#include <stdint.h>
#include <stddef.h>

// ---------------- problem constants (from reference) ----------------
#define MIN_N     2
#define MAX_N     5
#define KDRAFT    8
#define BATCH     256
#define MAX_LEN   32768

// ---------------- kernel tiling ----------------
#define THREADS            256                    // 8 waves (wave32)
#define TILE_TOKENS        4096                   // 16 KB per buffer
#define CHUNKS             (TILE_TOKENS / 4)      // 1024 x b128 chunks per tile
#define CHUNKS_PER_THREAD  (CHUNKS / THREADS)     // 4 async b128 issues / thread / tile

// Types matching the builtin's expected signature (from clang's diagnostic):
//   param0: v4i32 in address_space(1)  (global source, printed "__device__")
//   param1: v4i32 in address_space(3)  (LDS destination)
typedef int v4i __attribute__((vector_size(4 * sizeof(int))));
typedef __attribute__((address_space(1))) v4i g_v4i;
typedef __attribute__((address_space(3))) v4i l_v4i;
typedef __attribute__((address_space(3))) int lds_i32;

// ---- CDNA5 async global->LDS load (ASYNCcnt-tracked), builtin or inline asm ----
__device__ __forceinline__ void async_load_b128(const int* gsrc, int* lds_dst) {
#if __has_builtin(__builtin_amdgcn_global_load_async_to_lds_b128)
  __builtin_amdgcn_global_load_async_to_lds_b128(
      (g_v4i*)gsrc, (l_v4i*)lds_dst, /*offset=*/0, /*cpol=*/0);
#else
  unsigned lds_off = (unsigned)(size_t)(lds_i32*)lds_dst;     // raw LDS byte address
  asm volatile("global_load_async_to_lds_b128 %0, %1, off"
               :: "v"(lds_off), "v"((unsigned long long)(size_t)gsrc)
               : "memory");
#endif
}

#if __has_builtin(__builtin_amdgcn_s_wait_asynccnt)
#define WAIT_ASYNC(n) __builtin_amdgcn_s_wait_asynccnt(n)
#else
#define WAIT_ASYNC(n) asm volatile("s_wait_asynccnt %0" :: "i"(n) : "memory")
#endif

// One block per batch row. Streams the row through LDS with double-buffered
// async loads; each wave owns exactly the chunks it loaded, so the hot loop
// needs no workgroup barriers — only per-wave s_wait_asynccnt.
__global__ __launch_bounds__(THREADS)
void ngram_draft_kernel(const int* __restrict__ num_tokens,     // [B]
                        const int* __restrict__ token_ids,      // [B, MAX_LEN]
                        const unsigned char* __restrict__ mask, // [B] (bool)
                        int* __restrict__ out_tokens,           // [B, KDRAFT]
                        unsigned char* __restrict__ out_empty)  // [B] (bool)
{
  __shared__ int s_buf[2][TILE_TOKENS];     // 32 KB double buffer
  __shared__ int s_minp[MAX_N - MIN_N + 1]; // earliest match END position per n
  __shared__ int s_suf[MAX_N];              // trailing MAX_N tokens (suffix)

  const int b   = blockIdx.x;
  const int tid = threadIdx.x;
  const int len = num_tokens[b];            // guaranteed >= 64 by setup
  const int* g  = token_ids + (size_t)b * MAX_LEN;

  if (tid < MAX_N - MIN_N + 1) s_minp[tid] = 0x7fffffff;
  if (tid < MAX_N)             s_suf[tid]  = g[len - 1 - tid];
  __syncthreads();

  const int suf0 = s_suf[0], suf1 = s_suf[1], suf2 = s_suf[2],
            suf3 = s_suf[3], suf4 = s_suf[4];
  const int plimit = len - 2;               // window end p = j+n-1 must be <= len-2
  const int ntiles = (len + TILE_TOKENS - 1) / TILE_TOKENS;  // <= 8; fits MAX_LEN

  // Prefetch tile 0 (each thread issues CHUNKS_PER_THREAD b128 async loads).
  #pragma unroll
  for (int i = 0; i < CHUNKS_PER_THREAD; ++i) {
    const int c = tid + i * THREADS;
    async_load_b128(g + c * 4, &s_buf[0][c * 4]);
  }

  for (int t = 0; t < ntiles; ++t) {
    const int cur = t & 1;
    if (t + 1 < ntiles) {
      // Issue next tile while current is in flight, then wait so that only the
      // next tile's CHUNKS_PER_THREAD loads remain outstanding (=> tile t landed).
      const int base_n = (t + 1) * TILE_TOKENS;  // full-tile read stays within
      #pragma unroll                              // the [B, MAX_LEN] allocation
      for (int i = 0; i < CHUNKS_PER_THREAD; ++i) {
        const int c = tid + i * THREADS;
        async_load_b128(g + base_n + c * 4, &s_buf[cur ^ 1][c * 4]);
      }
      WAIT_ASYNC(CHUNKS_PER_THREAD);
    } else {
      WAIT_ASYNC(0);
    }

    // Scan tile t out of LDS. Thread reads back exactly the chunks it loaded.
    const int base = t * TILE_TOKENS;
    #pragma unroll
    for (int i = 0; i < CHUNKS_PER_THREAD; ++i) {
      const int c  = tid + i * THREADS;
      const int4 v = ((const int4*)s_buf[cur])[c];   // ds_load_b128
      const int p0 = base + c * 4;
      const int tk[4] = {v.x, v.y, v.z, v.w};
      #pragma unroll
      for (int e = 0; e < 4; ++e) {
        const int p = p0 + e;
        if (tk[e] == suf0 && p >= 1 && p <= plimit) {
          // Rare hit (~len/32000 per row): backward-extend against the suffix.
          int m = 1;
          if (g[p - 1] == suf1) {           m = 2;
            if (p >= 2 && g[p - 2] == suf2) { m = 3;
              if (p >= 3 && g[p - 3] == suf3) { m = 4;
                if (p >= 4 && g[p - 4] == suf4) { m = 5; } } } }
          #pragma unroll
          for (int n = MIN_N; n <= MAX_N; ++n)
            if (m >= n) atomicMin(&s_minp[n - MIN_N], p);
        }
      }
    }
    // No barrier needed: buffers and scan chunks are wave-private, and a
    // wave's next-iteration async issue cannot pass its own completed ds reads.
  }

  __syncthreads();   // publish s_minp atomics

  if (tid == 0) {
    // Prefer the LONGEST n that matched; earliest end position p gives
    // draft_start = p + 1 (== match_pos + n) for any n.
    int best_p = -1;
    #pragma unroll
    for (int n = MAX_N; n >= MIN_N; --n) {
      const int v = s_minp[n - MIN_N];
      if (best_p < 0 && v != 0x7fffffff) best_p = v;
    }
    const bool msk = (mask[b] != 0);
    const bool has = (best_p >= 0);
    const int draft_start = has ? best_p + 1 : 0;
    const int avail = len - draft_start;   // >= 1 whenever has==true

    bool any = false;
    #pragma unroll
    for (int k = 0; k < KDRAFT; ++k) {
      int idx = draft_start + k;
      if (idx > MAX_LEN - 1) idx = MAX_LEN - 1;       // clip like the reference
      const int tokv = g[idx];
      const int o = (has && msk && (k < avail)) ? tokv : -1;
      out_tokens[b * KDRAFT + k] = o;
      any = any || (o != -1);
    }
    out_empty[b] = any ? 0 : 1;
  }
}

extern "C" void kernel_launch(void* const* d_in, const int* in_sizes, int n_in,
                              void* d_out, int out_size, void* d_ws, size_t ws_size,
                              hipStream_t stream) {
  (void)in_sizes; (void)n_in; (void)out_size; (void)d_ws; (void)ws_size;
  const int*           num_tokens = (const int*)d_in[0];           // int32 [256]
  const int*           token_ids  = (const int*)d_in[1];           // int32 [256,32768]
  const unsigned char* cmask      = (const unsigned char*)d_in[2]; // bool  [256]

  // Tuple output concatenated flat in return order:
  //   draft_tokens int32 [256,8]  then  is_empty bool [256]
  int*           out_tokens = (int*)d_out;
  unsigned char* out_empty  = (unsigned char*)d_out + (size_t)BATCH * KDRAFT * sizeof(int);

  ngram_draft_kernel<<<BATCH, THREADS, 0, stream>>>(
      num_tokens, token_ids, cmask, out_tokens, out_empty);
}